// _RpnLabelEncoder_11476152615315
// MI455X (gfx1250) — compile-verified
//
#include <hip/hip_runtime.h>
#include <hip/hip_bf16.h>
#include <stdint.h>

// ---------------------------------------------------------------------------
// RPN label encoder for MI455X (gfx1250, wave32).
//   out layout (float32): [enc N*4][box_w N][class_t N][class_sw N]
//   ws  layout:           [flags u8 N][keys u32 N][hist u32 512][SelState]
// ---------------------------------------------------------------------------

#define MAXG   128
#define KSEL   128u   // 256 samples * 0.5 positive fraction = 128 pos, 128 neg

typedef int v4i_cdna5 __attribute__((ext_vector_type(4)));
#define AS_GLOBAL __attribute__((address_space(1)))
#define AS_LDS    __attribute__((address_space(3)))

struct U2 { unsigned x, y; };

// threefry2x32-20 (JAX PRNG). Rotations {13,15,26,6} / {17,29,16,24}.
__host__ __device__ inline unsigned rotl32(unsigned v, unsigned r) {
  return (v << r) | (v >> (32u - r));
}

__host__ __device__ inline U2 threefry2x32(unsigned k0, unsigned k1,
                                           unsigned c0, unsigned c1) {
  const unsigned ks2 = k0 ^ k1 ^ 0x1BD11BDAu;
  unsigned ks[3] = {k0, k1, ks2};
  unsigned x0 = c0 + k0;
  unsigned x1 = c1 + k1;
  const unsigned RA[4] = {13u, 15u, 26u, 6u};
  const unsigned RB[4] = {17u, 29u, 16u, 24u};
#pragma unroll
  for (int blk = 0; blk < 5; ++blk) {
    const unsigned* R = (blk & 1) ? RB : RA;
#pragma unroll
    for (int r = 0; r < 4; ++r) {
      x0 += x1;
      x1 = rotl32(x1, R[r]);
      x1 ^= x0;
    }
    x0 += ks[(blk + 1) % 3];
    x1 += ks[(blk + 2) % 3] + (unsigned)(blk + 1);
  }
  return {x0, x1};
}

// JAX random_bits(key,(N,)) counter layout: i<N/2 -> tf(i, i+N/2).x ; else .y
__device__ inline unsigned jax_bits(unsigned k0, unsigned k1, int i, int n) {
  const int half = n >> 1;
  if (i < half) return threefry2x32(k0, k1, (unsigned)i, (unsigned)(i + half)).x;
  return threefry2x32(k0, k1, (unsigned)(i - half), (unsigned)i).y;
}

struct SelState {
  unsigned prefix[2];    // decided high bits of the k-th largest key
  unsigned k[2];         // remaining rank within current prefix bucket
  unsigned take_all[2];  // fewer candidates than k -> select every flagged one
};

// ---------------------------------------------------------------------------
// Main kernel: IoU argmax (division-free), delta encoding, flags + PRNG keys.
// GT boxes staged into LDS through the CDNA5 async global->LDS path.
// ---------------------------------------------------------------------------
__global__ void rpn_main_kernel(const float* __restrict__ anchors,
                                const float* __restrict__ gt_boxes,
                                float* __restrict__ enc,
                                float* __restrict__ box_w,
                                float* __restrict__ class_t,
                                unsigned char* __restrict__ flags,
                                unsigned* __restrict__ keys,
                                int N, int G,
                                unsigned kp0, unsigned kp1,
                                unsigned kn0, unsigned kn1) {
  __shared__ float s_gt[MAXG * 4];
  __shared__ float s_area[MAXG];
  const int tid = threadIdx.x;
  const int g = (G < MAXG) ? G : MAXG;

#if defined(__gfx1250__)
  // --- async copy of the 16B/box GT tile straight into LDS (ASYNCcnt path)
  if (tid < g) {
#if __has_builtin(__builtin_amdgcn_global_load_async_to_lds_b128)
    // signature (from clang diagnostic): (v4i AS1* src, v4i AS3* dst,
    //                                     imm i32 offset, imm i32 cpol)
    __builtin_amdgcn_global_load_async_to_lds_b128(
        (v4i_cdna5 AS_GLOBAL*)(gt_boxes + 4 * tid),
        (v4i_cdna5 AS_LDS*)(&s_gt[4 * tid]),
        0, 0);
#else
    unsigned lds_off = (unsigned)(unsigned long long)(
        (float AS_LDS*)(&s_gt[4 * tid]));
    const float* src = gt_boxes + 4 * tid;
    asm volatile("global_load_async_to_lds_b128 %0, %1, off"
                 :: "v"(lds_off), "v"(src) : "memory");
#endif
  }
#if __has_builtin(__builtin_amdgcn_s_wait_asynccnt)
  __builtin_amdgcn_s_wait_asynccnt(0);
#else
  asm volatile("s_wait_asynccnt 0" ::: "memory");
#endif
#else
  for (int t = tid; t < g * 4; t += blockDim.x) s_gt[t] = gt_boxes[t];
#endif
  __syncthreads();

  if (tid < g) {
    const float y0 = s_gt[4 * tid + 0], x0 = s_gt[4 * tid + 1];
    const float y1 = s_gt[4 * tid + 2], x1 = s_gt[4 * tid + 3];
    s_area[tid] = (y1 - y0) * (x1 - x0);
  }
  __syncthreads();

  const int i = blockIdx.x * blockDim.x + tid;
  if (i >= N) return;

  const float4 a = ((const float4*)anchors)[i];  // coalesced B128 load
  const float ay0 = a.x, ax0 = a.y, ay1 = a.z, ax1 = a.w;
  const float areaA = (ay1 - ay0) * (ax1 - ax0);

  // Division-free argmax over IoU: iou_j > iou_best <=> inter_j*bu > bi*uni_j
  float bi = 0.0f, bu = 1.0f;
  int bj = 0;
  for (int j = 0; j < g; ++j) {
    const float gy0 = s_gt[4 * j + 0], gx0 = s_gt[4 * j + 1];
    const float gy1 = s_gt[4 * j + 2], gx1 = s_gt[4 * j + 3];
    float ih = fminf(ay1, gy1) - fmaxf(ay0, gy0);
    float iw = fminf(ax1, gx1) - fmaxf(ax0, gx0);
    ih = fmaxf(ih, 0.0f);
    iw = fmaxf(iw, 0.0f);
    const float inter = ih * iw;
    const float uni = areaA + s_area[j] - inter;
    if (inter * bu > bi * uni) { bi = inter; bu = uni; bj = j; }
  }
  const float maxval = (bu > 0.0f) ? (bi / bu) : 0.0f;
  const bool pos = maxval >= 0.7f;
  const bool neg = maxval < 0.3f;

  // Delta encoding vs matched GT (variance (0.1,0.1,0.2,0.2) -> *10,*10,*5,*5)
  const float gy0 = s_gt[4 * bj + 0], gx0 = s_gt[4 * bj + 1];
  const float gy1 = s_gt[4 * bj + 2], gx1 = s_gt[4 * bj + 3];
  const float ah = ay1 - ay0, aw = ax1 - ax0;
  const float acy = ay0 + 0.5f * ah, acx = ax0 + 0.5f * aw;
  const float gh = gy1 - gy0, gw = gx1 - gx0;
  const float gcy = gy0 + 0.5f * gh, gcx = gx0 + 0.5f * gw;
  float4 d;
  d.x = 10.0f * (gcy - acy) / ah;
  d.y = 10.0f * (gcx - acx) / aw;
  d.z = 5.0f * __logf(gh / ah);
  d.w = 5.0f * __logf(gw / aw);
  ((float4*)enc)[i] = d;  // coalesced B128 store

  const float posf = pos ? 1.0f : 0.0f;
  box_w[i] = posf;
  class_t[i] = posf;

  const unsigned f = pos ? 1u : (neg ? 2u : 0u);
  flags[i] = (unsigned char)f;
  unsigned bits = 0u;
  if (f) {
    const unsigned k0 = pos ? kp0 : kn0;
    const unsigned k1 = pos ? kp1 : kn1;
    bits = jax_bits(k0, k1, i, N);  // monotone key for uniform score
  }
  keys[i] = bits;
}

// ---------------------------------------------------------------------------
// Radix-select (top-128 per class) over the flagged PRNG keys.
// ---------------------------------------------------------------------------
__global__ void sel_init_kernel(unsigned* hist, SelState* st) {
  const int t = blockIdx.x * blockDim.x + threadIdx.x;
  if (t < 512) hist[t] = 0u;
  if (t == 0) {
    st->prefix[0] = 0u;  st->prefix[1] = 0u;
    st->k[0] = KSEL;     st->k[1] = KSEL;
    st->take_all[0] = 0u; st->take_all[1] = 0u;
  }
}

__global__ void sel_hist_kernel(const unsigned char* __restrict__ flags,
                                const unsigned* __restrict__ keys,
                                unsigned* __restrict__ hist,
                                const SelState* __restrict__ st,
                                int N, int shift) {
  __shared__ unsigned lh[512];
  for (int t = threadIdx.x; t < 512; t += blockDim.x) lh[t] = 0u;
  __syncthreads();

  const unsigned pre0 = st->prefix[0], pre1 = st->prefix[1];
  const unsigned ta0 = st->take_all[0], ta1 = st->take_all[1];
  const unsigned himask = (shift + 8 >= 32) ? 0u : ~((1u << (shift + 8)) - 1u);

  const int stride = gridDim.x * blockDim.x;
  for (int i = blockIdx.x * blockDim.x + threadIdx.x; i < N; i += stride) {
    if (i + stride < N) {
      __builtin_prefetch(flags + i + stride, 0, 0);  // global_prefetch_b8
      __builtin_prefetch(keys + i + stride, 0, 0);
    }
    const unsigned f = flags[i];
    if (f == 0u) continue;
    const unsigned c = f - 1u;
    if (c ? ta1 : ta0) continue;
    const unsigned key = keys[i];
    const unsigned pre = c ? pre1 : pre0;
    if ((key & himask) != (pre & himask)) continue;
    atomicAdd(&lh[c * 256u + ((key >> shift) & 255u)], 1u);
  }
  __syncthreads();
  for (int t = threadIdx.x; t < 512; t += blockDim.x)
    if (lh[t]) atomicAdd(&hist[t], lh[t]);
}

__global__ void sel_scan_kernel(unsigned* hist, SelState* st, int shift) {
  if (threadIdx.x != 0 || blockIdx.x != 0) return;
  for (int c = 0; c < 2; ++c) {
    if (!st->take_all[c]) {
      const unsigned k = st->k[c];
      unsigned long long cum = 0ull;
      int found = -1;
      for (int b = 255; b >= 0; --b) {
        const unsigned cnt = hist[c * 256 + b];
        cum += cnt;
        if (cum >= k) {
          found = b;
          st->k[c] = k - (unsigned)(cum - cnt);
          st->prefix[c] |= ((unsigned)b) << shift;
          break;
        }
      }
      if (found < 0) st->take_all[c] = 1u;  // fewer candidates than k
    }
    for (int b = 0; b < 256; ++b) hist[c * 256 + b] = 0u;  // next pass
  }
}

__global__ void sel_final_kernel(const unsigned char* __restrict__ flags,
                                 const unsigned* __restrict__ keys,
                                 const SelState* __restrict__ st,
                                 float* __restrict__ class_sw, int N) {
  const int i = blockIdx.x * blockDim.x + threadIdx.x;
  if (i >= N) return;
  const unsigned f = flags[i];
  float v = 0.0f;
  if (f) {
    const unsigned c = f - 1u;
    if (st->take_all[c] || keys[i] >= st->prefix[c]) v = 1.0f;
  }
  class_sw[i] = v;
}

// ---------------------------------------------------------------------------
extern "C" void kernel_launch(void* const* d_in, const int* in_sizes, int n_in,
                              void* d_out, int out_size, void* d_ws,
                              size_t ws_size, hipStream_t stream) {
  const int N = in_sizes[0] / 4;   // 262144 anchors
  const int G = in_sizes[1] / 4;   // 128 gt boxes
  const float* anchors = (const float*)d_in[0];
  const float* gt = (const float*)d_in[1];

  float* out = (float*)d_out;
  float* enc      = out;
  float* box_w    = out + 4 * (size_t)N;
  float* class_t  = out + 5 * (size_t)N;
  float* class_sw = out + 6 * (size_t)N;

  unsigned char* flags = (unsigned char*)d_ws;
  const size_t flags_bytes = ((size_t)N + 255) & ~(size_t)255;
  unsigned* keys = (unsigned*)((char*)d_ws + flags_bytes);
  unsigned* hist = keys + N;
  SelState* st = (SelState*)(hist + 512);

  // Host-side JAX key derivation: key(42) = (0,42); split -> kp, kn.
  // bits = tf(key,[0,1,2,3]) with split-counter layout:
  //   kp = (tf(0,2).x, tf(1,3).x), kn = (tf(0,2).y, tf(1,3).y)
  const U2 pa = threefry2x32(0u, 42u, 0u, 2u);
  const U2 pb = threefry2x32(0u, 42u, 1u, 3u);
  const unsigned kp0 = pa.x, kp1 = pb.x;
  const unsigned kn0 = pa.y, kn1 = pb.y;

  const int threads = 256;  // 8 wave32 waves
  const int blocks = (N + threads - 1) / threads;

  rpn_main_kernel<<<blocks, threads, 0, stream>>>(
      anchors, gt, enc, box_w, class_t, flags, keys, N, G,
      kp0, kp1, kn0, kn1);

  sel_init_kernel<<<2, threads, 0, stream>>>(hist, st);
  for (int shift = 24; shift >= 0; shift -= 8) {
    sel_hist_kernel<<<512, threads, 0, stream>>>(flags, keys, hist, st, N, shift);
    sel_scan_kernel<<<1, 1, 0, stream>>>(hist, st, shift);
  }
  sel_final_kernel<<<blocks, threads, 0, stream>>>(flags, keys, st, class_sw, N);
}